// ComplExDecoder_30674656428512
// MI455X (gfx1250) — compile-verified
//
#include <hip/hip_runtime.h>
#include <math.h>

typedef __attribute__((ext_vector_type(2))) float v2f;
typedef __attribute__((ext_vector_type(8))) float v8f;

#define HIDDEN 512
#define HALF_H 256

// ---------------------------------------------------------------------------
// Kernel A: per-node inverse L2 norm.  One wave per node, 4x b128 coalesced
// loads per lane, wave32 xor-shuffle reduction.  ~205 MB read -> ~9 us.
// ---------------------------------------------------------------------------
__global__ __launch_bounds__(256) void node_inv_norm_kernel(
    const float* __restrict__ z, float* __restrict__ inv_norm, int num_nodes) {
  const int lane = threadIdx.x & 31;
  const int node = blockIdx.x * (blockDim.x >> 5) + (threadIdx.x >> 5);
  if (node >= num_nodes) return;
  const float* row = z + (size_t)node * HIDDEN;
  float ss = 0.f;
#pragma unroll
  for (int j = 0; j < 4; ++j) {
    const float4 v = *(const float4*)(row + lane * 4 + j * 128);
    ss += v.x * v.x + v.y * v.y + v.z * v.z + v.w * v.w;
  }
#pragma unroll
  for (int off = 16; off > 0; off >>= 1) ss += __shfl_xor(ss, off, 32);
  if (lane == 0) {
    const float n = fmaxf(sqrtf(ss), 1e-12f);
    inv_norm[node] = 1.0f / n;
  }
}

// ---------------------------------------------------------------------------
// Kernel B: 16 edges per wave, mapped onto the V_WMMA_F32_16X16X4_F32
// A-matrix layout: lane l -> edge (l & 15), K = 2*(l>>4) + {0,1}.
// Per K-block of 4 hidden indices, each lane computes its two f32 terms
//   t = inv_ns*inv_nd * ( rre*(sre*dre + sim*dim) + rim*(sre*dim - sim*dre) )
// and the WMMA against an all-ones B matrix accumulates the 16 per-edge sums
// in the f32 D accumulator (every N column holds an identical copy).
// ---------------------------------------------------------------------------
__global__ __launch_bounds__(256) void complex_score_wmma_kernel(
    const float* __restrict__ z,
    const int* __restrict__ edge_index,   // [2, E] row-major
    const int* __restrict__ edge_type,    // [E]
    const float* __restrict__ rel_re,     // [R, 256]
    const float* __restrict__ rel_im,     // [R, 256]
    const float* __restrict__ inv_norm,   // [N]
    float* __restrict__ out,              // [E]
    int num_edges) {
  const int lane  = threadIdx.x & 31;
  const int wave  = (blockIdx.x * blockDim.x + threadIdx.x) >> 5;
  const int ebase = wave * 16;
  const int m     = lane & 15;   // A-matrix row (edge within tile)
  const int half  = lane >> 4;   // K half: K = 2*half + {0,1}

  int e = ebase + m;
  e = (e < num_edges) ? e : (num_edges - 1);   // clamp: EXEC stays all-ones

  const int src = edge_index[e];
  const int dst = edge_index[num_edges + e];
  const int typ = edge_type[e];
  const float s = inv_norm[src] * inv_norm[dst];

  const float* zs = z + (size_t)src * HIDDEN;
  const float* zd = z + (size_t)dst * HIDDEN;
  const float* rr = rel_re + (size_t)typ * HALF_H;
  const float* ri = rel_im + (size_t)typ * HALF_H;
  const int hb = half * 2;

  v8f acc = {0.f, 0.f, 0.f, 0.f, 0.f, 0.f, 0.f, 0.f};
  v2f ones;
  ones.x = 1.0f;
  ones.y = 1.0f;

#pragma unroll 8
  for (int kb = 0; kb < HALF_H / 4; ++kb) {
    const int h = kb * 4 + hb;
    const v2f sre = *(const v2f*)(zs + h);
    const v2f sim = *(const v2f*)(zs + h + HALF_H);
    const v2f dre = *(const v2f*)(zd + h);
    const v2f dim = *(const v2f*)(zd + h + HALF_H);
    const v2f wre = *(const v2f*)(rr + h);
    const v2f wim = *(const v2f*)(ri + h);

    v2f a;
    {
      const float p0 = sre.x * dre.x + sim.x * dim.x;
      const float q0 = sre.x * dim.x - sim.x * dre.x;
      a.x = (wre.x * p0 + wim.x * q0) * s;
      const float p1 = sre.y * dre.y + sim.y * dim.y;
      const float q1 = sre.y * dim.y - sim.y * dre.y;
      a.y = (wre.y * p1 + wim.y * q1) * s;
    }
    // D[m][n] += sum_k A[m][k]  (B = all ones), f32-exact accumulation
    acc = __builtin_amdgcn_wmma_f32_16x16x4_f32(
        /*neg_a=*/false, a, /*neg_b=*/false, ones,
        /*c_mod=*/(short)0, acc, /*reuse_a=*/false, /*reuse_b=*/false);
  }

  // D layout: VGPR v, lanes 0-15 -> M=v, lanes 16-31 -> M=v+8 (all N columns
  // identical because B was all-ones).  Lane 0 writes edges 0..7 of the tile,
  // lane 16 writes edges 8..15.
  if (m == 0) {
#pragma unroll
    for (int v = 0; v < 8; ++v) {
      const int ee = ebase + half * 8 + v;
      if (ee < num_edges) out[ee] = acc[v];
    }
  }
}

// ---------------------------------------------------------------------------
extern "C" void kernel_launch(void* const* d_in, const int* in_sizes, int n_in,
                              void* d_out, int out_size, void* d_ws, size_t ws_size,
                              hipStream_t stream) {
  const float* z          = (const float*)d_in[0];
  const int*   edge_index = (const int*)d_in[1];
  const int*   edge_type  = (const int*)d_in[2];
  const float* rel_re     = (const float*)d_in[3];
  const float* rel_im     = (const float*)d_in[4];
  float* out      = (float*)d_out;
  float* inv_norm = (float*)d_ws;   // 400 KB scratch

  const int num_nodes = in_sizes[0] / HIDDEN;
  const int num_edges = in_sizes[2];

  {
    const int waves_per_block = 256 / 32;
    const int blocks = (num_nodes + waves_per_block - 1) / waves_per_block;
    node_inv_norm_kernel<<<blocks, 256, 0, stream>>>(z, inv_norm, num_nodes);
  }
  {
    const int tiles = (num_edges + 15) / 16;        // one wave per 16 edges
    const int waves_per_block = 256 / 32;
    const int blocks = (tiles + waves_per_block - 1) / waves_per_block;
    complex_score_wmma_kernel<<<blocks, 256, 0, stream>>>(
        z, edge_index, edge_type, rel_re, rel_im, inv_norm, out, num_edges);
  }
}